// FusedLoRA_19516331393051
// MI455X (gfx1250) — compile-verified
//
#include <hip/hip_runtime.h>

// ---------------------------------------------------------------------------
// Fused 3-way LoRA: out[m, l*4096+n] = sum_r ( sum_k x[m,k]*A_l[k,r] ) * B_l[r,n]
// x: [16384, 4096] f32, A_l: [4096, 8] f32, B_l: [8, 4096] f32, out: [16384,12288] f32
//
// Bandwidth-bound (256 MB x read + 768 MB out write vs 6.4 GFLOP), so a single
// fused kernel: each WG owns a 16-row M-tile, computes xa[16x24] once via
// V_WMMA_F32_16X16X4_F32 (8 waves split K, x double-buffered through LDS),
// then expands rank-8 to the 768 KB output tile with nontemporal b128 stores.
// ---------------------------------------------------------------------------

typedef float v2f __attribute__((ext_vector_type(2)));
typedef float v4f __attribute__((ext_vector_type(4)));
typedef float v8f __attribute__((ext_vector_type(8)));

#define DIM      4096
#define OUTD     12288
#define MTILE    16
#define KCHUNK   256
#define XPITCH   260   // 256 + 4 pad -> bank = (4*row + k) % 64, conflict-free b64 reads
#define XAPITCH  32    // 24 cols padded to 32 -> 128B rows

__global__ __launch_bounds__(256)
void lora_fused_kernel(const float* __restrict__ x,
                       const float* __restrict__ A0, const float* __restrict__ B0,
                       const float* __restrict__ A1, const float* __restrict__ B1,
                       const float* __restrict__ A2, const float* __restrict__ B2,
                       float* __restrict__ out)
{
    __shared__ float lds_x[2][MTILE * XPITCH]; // 2 x 16640 B : double-buffered x chunks
    __shared__ float lds_red[8 * 512];         // 16384 B     : per-wave WMMA partials
    __shared__ float lds_xa[MTILE * XAPITCH];  //  2048 B     : xa[16][24] (+pad)

    const int tid  = threadIdx.x;
    const int lane = tid & 31;
    const int wave = tid >> 5;
    const int m0   = blockIdx.x * MTILE;

    // ---------------- Phase 1: xa = x_tile @ [A0 | A1 | A2] ----------------
    // N-tile0 cols 0..15  = lora0 r0-7 | lora1 r0-7
    // N-tile1 cols 16..31 = lora2 r0-7 | zero pad
    v8f acc0 = {0.f,0.f,0.f,0.f,0.f,0.f,0.f,0.f};
    v8f acc1 = {0.f,0.f,0.f,0.f,0.f,0.f,0.f,0.f};

    const int  xrow   = tid >> 4;            // cooperative-load row    0..15
    const int  xf4    = tid & 15;            // cooperative-load f4 col 0..15
    const int  arow   = lane & 15;           // WMMA A-operand M index
    const int  ksub   = (lane < 16) ? 0 : 2; // A/B K sub-offset per half-wave
    const int  bn     = lane & 15;           // WMMA B-operand N index
    const int  r8     = bn & 7;
    const bool hizero = (bn >= 8);           // N-tile1 cols 8..15 are zero pad
    const float* Alo  = hizero ? A1 : A0;    // uniform cndmask, no branch

    const float* srcrow = x + (size_t)(m0 + xrow) * DIM + xf4 * 4;

    // prologue: stage chunk 0 into buffer 0
    v4f xr[4];
    #pragma unroll
    for (int j = 0; j < 4; ++j)
        xr[j] = __builtin_nontemporal_load((const v4f*)(srcrow + j * 64));
    {
        float* dst = &lds_x[0][xrow * XPITCH + xf4 * 4];
        #pragma unroll
        for (int j = 0; j < 4; ++j) {
            dst[j*64 + 0] = xr[j].x; dst[j*64 + 1] = xr[j].y;
            dst[j*64 + 2] = xr[j].z; dst[j*64 + 3] = xr[j].w;
        }
    }

    int cur = 0;
    for (int kc = 0; kc < DIM; kc += KCHUNK) {
        __syncthreads();                      // buf[cur] ready; buf[cur^1] free
        const bool more = (kc + KCHUNK) < DIM;

        // issue next chunk's global loads before compute (latency overlap)
        if (more) {
            const float* srcn = srcrow + kc + KCHUNK;
            #pragma unroll
            for (int j = 0; j < 4; ++j)
                xr[j] = __builtin_nontemporal_load((const v4f*)(srcn + j * 64));
        }

        // wave w handles k in [kc + 32w, kc + 32w + 32): 8 WMMA k-steps
        const int klo = wave * 32;
        const float* lx = &lds_x[cur][arow * XPITCH + ksub];
        #pragma unroll
        for (int s = 0; s < 8; ++s) {
            const int k  = klo + s * 4;        // local k (multiple of 4)
            const int kg = kc + k + ksub;      // global k for this lane's B rows
            // A operand: x[arow][k+ksub .. +1]  (ds_load_b64, conflict-free)
            v2f a = *(const v2f*)(lx + k);
            // B operand rows kg, kg+1 — all loads unconditional (addresses are
            // in-bounds for every lane); zero pad via value select (v_cndmask)
            v2f b0, b1;
            b0.x = Alo[(size_t)kg * 8 + r8];
            b0.y = Alo[(size_t)(kg + 1) * 8 + r8];
            const float a2lo = A2[(size_t)kg * 8 + r8];
            const float a2hi = A2[(size_t)(kg + 1) * 8 + r8];
            b1.x = hizero ? 0.f : a2lo;
            b1.y = hizero ? 0.f : a2hi;
            acc0 = __builtin_amdgcn_wmma_f32_16x16x4_f32(false, a, false, b0,
                                                         (short)0, acc0, false, false);
            acc1 = __builtin_amdgcn_wmma_f32_16x16x4_f32(false, a, false, b1,
                                                         (short)0, acc1, false, false);
        }

        // stage next chunk into the other buffer (after compute issued)
        if (more) {
            float* dst = &lds_x[cur ^ 1][xrow * XPITCH + xf4 * 4];
            #pragma unroll
            for (int j = 0; j < 4; ++j) {
                dst[j*64 + 0] = xr[j].x; dst[j*64 + 1] = xr[j].y;
                dst[j*64 + 2] = xr[j].z; dst[j*64 + 3] = xr[j].w;
            }
        }
        cur ^= 1;
    }

    // cross-wave reduction of WMMA partials through LDS
    #pragma unroll
    for (int v = 0; v < 8; ++v) {
        lds_red[wave * 512 +       v * 32 + lane] = acc0[v];
        lds_red[wave * 512 + 256 + v * 32 + lane] = acc1[v];
    }
    __syncthreads();

    #pragma unroll
    for (int t = 0; t < 2; ++t) {
        const int idx = tid + t * 256;          // 0..511
        float s = 0.f;
        #pragma unroll
        for (int w = 0; w < 8; ++w) s += lds_red[w * 512 + idx];
        // C/D layout: VGPR v -> M = v (+8 for lanes 16..31), N = lane%16
        const int tile = idx >> 8;
        const int v    = (idx >> 5) & 7;
        const int ln   = idx & 31;
        const int row  = v + ((ln & 16) ? 8 : 0);
        const int col  = tile * 16 + (ln & 15);
        if (col < 24) lds_xa[row * XAPITCH + col] = s;
    }
    __syncthreads();

    // ---------------- Phase 2: out_tile = xa @ [B0 ; B1 ; B2] ----------------
    const int orow = tid >> 4;   // 0..15
    const int cg   = tid & 15;   // 0..15 -> col = cg*4 + 64*j (256B/row per wave)

    float xar[24];
    #pragma unroll
    for (int i = 0; i < 24; ++i) xar[i] = lds_xa[orow * XAPITCH + i];

    float* orow_ptr = out + (size_t)(m0 + orow) * OUTD;
    const float* Bmats[3] = {B0, B1, B2};

    #pragma unroll
    for (int l = 0; l < 3; ++l) {
        const float* Bl = Bmats[l];
        float* op = orow_ptr + l * DIM;
        #pragma unroll 2
        for (int j = 0; j < 64; ++j) {
            const int col = cg * 4 + j * 64;
            v4f acc = {0.f, 0.f, 0.f, 0.f};
            #pragma unroll
            for (int r = 0; r < 8; ++r) {
                v4f b = *(const v4f*)(Bl + (size_t)r * DIM + col);  // L2-hot (384 KB)
                const float xv = xar[l * 8 + r];
                acc.x += xv * b.x; acc.y += xv * b.y;
                acc.z += xv * b.z; acc.w += xv * b.w;
            }
            __builtin_nontemporal_store(acc, (v4f*)(op + col));     // streamed 768 MB
        }
    }
}

extern "C" void kernel_launch(void* const* d_in, const int* in_sizes, int n_in,
                              void* d_out, int out_size, void* d_ws, size_t ws_size,
                              hipStream_t stream) {
    const float* x  = (const float*)d_in[0];
    const float* A0 = (const float*)d_in[1];
    const float* B0 = (const float*)d_in[2];
    const float* A1 = (const float*)d_in[3];
    const float* B1 = (const float*)d_in[4];
    const float* A2 = (const float*)d_in[5];
    const float* B2 = (const float*)d_in[6];
    float* out = (float*)d_out;

    const int rows = 4 * 4096;            // 16384 flattened (batch, seq) rows
    dim3 grid(rows / MTILE);              // 1024 workgroups
    lora_fused_kernel<<<grid, 256, 0, stream>>>(x, A0, B0, A1, B1, A2, B2, out);
}